// EdgePooling_38654705664707
// MI455X (gfx1250) — compile-verified
//
#include <hip/hip_runtime.h>
#include <stdint.h>

#define CHN 128
#define ROUNDS 48

typedef __attribute__((ext_vector_type(2))) float v2f;
typedef __attribute__((ext_vector_type(8))) float v8f;

// ---- ordered-float helpers (monotonic uint encoding, for atomicMax on floats) ----
__device__ __forceinline__ unsigned f32_ordered(float f) {
  unsigned u = __float_as_uint(f);
  return (u & 0x80000000u) ? ~u : (u | 0x80000000u);
}
__device__ __forceinline__ float ordered_f32(unsigned u) {
  unsigned b = (u & 0x80000000u) ? (u & 0x7FFFFFFFu) : ~u;
  return __uint_as_float(b);
}

// ================= 1) node partial scores via V_WMMA_F32_16X16X4_F32 =================
// s_lo[v] = dot(x[v], w[0:128]), s_hi[v] = dot(x[v], w[128:256]).
// One wave computes a 16-node tile: D[16x16] = sum_k A[16x4] * B[4x16], B has 2 live cols.
// Weights staged in LDS; B fragment built branch-free (cndmask pointer + mask multiply)
// so EXEC stays full and no divergent save/restore appears around the WMMA stream.
__global__ void __launch_bounds__(256)
node_scores_wmma(const float* __restrict__ x, const float* __restrict__ lin_w,
                 float* __restrict__ s_lo, float* __restrict__ s_hi, int n_tiles) {
  __shared__ float wsh[2 * CHN];
  wsh[threadIdx.x] = lin_w[threadIdx.x];          // 256 threads load 256 weights
  __syncthreads();

  const int lane = threadIdx.x & 31;
  const int wave = threadIdx.x >> 5;
  const int tile = blockIdx.x * 8 + wave;
  if (tile >= n_tiles) return;                    // wave-uniform exit, EXEC stays full

  const int m  = tile * 16 + (lane & 15);         // A row for this lane
  const int k0 = (lane >> 4) * 2;                 // lanes 0-15: K=0,1 ; lanes 16-31: K=2,3
  const int nn = lane & 15;                       // B/D column
  const float* xr = x + (size_t)m * CHN;
  const float* wp = wsh + ((nn == 1) ? CHN : 0);  // cndmask on address, not a branch
  const float msk = (nn < 2) ? 1.0f : 0.0f;       // columns >= 2 contribute zero

  v8f acc = {};
#pragma unroll
  for (int kk = 0; kk < CHN; kk += 4) {
    v2f a = *(const v2f*)(xr + kk + k0);          // global_load_b64 (A fragment)
    v2f b = *(const v2f*)(wp + kk + k0);          // ds_load_b64 (B fragment)
    b.x *= msk;
    b.y *= msk;
    acc = __builtin_amdgcn_wmma_f32_16x16x4_f32(false, a, false, b,
                                                (short)0, acc, false, false);
  }
  // D layout: lane covers column nn; VGPR v holds row v + 8*(lane>>4)
  if (nn < 2) {
    float* dst = (nn == 0) ? s_lo : s_hi;
    int mb = tile * 16 + 8 * (lane >> 4);
#pragma unroll
    for (int v = 0; v < 8; ++v) dst[mb + v] = acc[v];
  }
}

__global__ void node_scores_tail(const float* __restrict__ x, const float* __restrict__ lin_w,
                                 float* s_lo, float* s_hi, int start, int N) {
  int v = start + blockIdx.x * 256 + threadIdx.x;
  if (v >= N) return;
  float a = 0.f, b = 0.f;
  for (int k = 0; k < CHN; ++k) {
    float xv = x[(size_t)v * CHN + k];
    a += xv * lin_w[k];
    b += xv * lin_w[CHN + k];
  }
  s_lo[v] = a; s_hi[v] = b;
}

// ================= 2) per-node init =================
__global__ void init_nodes(unsigned* nodeMax, float* den, int* used,
                           int* cluster, unsigned* inSet, int N) {
  int v = blockIdx.x * 256 + threadIdx.x;
  if (v >= N) return;
  nodeMax[v] = 0u;          // ordered encoding: 0 == most negative
  den[v] = 0.f;
  used[v] = 0;
  cluster[v] = v;
  inSet[v] = 1u;
}

// ================= 3) edge score pipeline (raw -> segmax -> exp/sum -> softmax) =====
__global__ void edge_raw_kernel(const float* __restrict__ s_lo, const float* __restrict__ s_hi,
                                const float* __restrict__ lin_b,
                                const int* __restrict__ row, const int* __restrict__ col,
                                float* raw, unsigned* nodeMax, int E) {
  int e = blockIdx.x * 256 + threadIdx.x;
  if (e >= E) return;
  float v = s_lo[row[e]] + s_hi[col[e]] + lin_b[0];
  raw[e] = v;
  atomicMax(&nodeMax[col[e]], f32_ordered(v));
}

__global__ void edge_exp_kernel(const float* __restrict__ raw, const int* __restrict__ col,
                                const unsigned* __restrict__ nodeMax,
                                float* ex, float* den, int E) {
  int e = blockIdx.x * 256 + threadIdx.x;
  if (e >= E) return;
  float m = ordered_f32(nodeMax[col[e]]);
  float v = expf(raw[e] - m);
  ex[e] = v;
  atomicAdd(&den[col[e]], v);
}

__global__ void edge_score_kernel(const float* __restrict__ ex, const float* __restrict__ den,
                                  const int* __restrict__ col, float* score,
                                  unsigned long long* ekey, int* match, int E) {
  int e = blockIdx.x * 256 + threadIdx.x;
  if (e >= E) return;
  float s = ex[e] / (den[col[e]] + 1e-16f) + 0.5f;
  score[e] = s;
  match[e] = 0;
  // key reproduces argsort(-score) stable order: high score wins, then low edge id
  ekey[e] = ((unsigned long long)f32_ordered(s) << 32) |
            (unsigned long long)(0xFFFFFFFFu - (unsigned)e);
}

// ================= 4) locally-dominant matching rounds (== sequential greedy) =======
__global__ void reset_keys(unsigned long long* nodeKey, int N) {
  int v = blockIdx.x * 256 + threadIdx.x;
  if (v < N) nodeKey[v] = 0ull;
}
__global__ void match_propose(const int* __restrict__ row, const int* __restrict__ col,
                              const int* __restrict__ used,
                              const unsigned long long* __restrict__ ekey,
                              unsigned long long* nodeKey, int E) {
  int e = blockIdx.x * 256 + threadIdx.x;
  if (e >= E) return;
  int r = row[e], c = col[e];
  if (used[r] | used[c]) return;
  unsigned long long k = ekey[e];
  atomicMax(&nodeKey[r], k);
  atomicMax(&nodeKey[c], k);
}
__global__ void match_accept(const int* __restrict__ row, const int* __restrict__ col,
                             int* used, const unsigned long long* __restrict__ ekey,
                             const unsigned long long* __restrict__ nodeKey,
                             int* match, int* cluster, unsigned* inSet, int E) {
  int e = blockIdx.x * 256 + threadIdx.x;
  if (e >= E) return;
  int r = row[e], c = col[e];
  if (used[r] | used[c]) return;
  unsigned long long k = ekey[e];
  if (nodeKey[r] == k && nodeKey[c] == k) {   // dominant at both endpoints
    match[e] = 1;
    used[r] = 1; used[c] = 1;                 // winners are vertex-disjoint: no races
    if (c != r) { cluster[c] = r; inSet[c] = 0u; }
  }
}

// ================= 5) single-block chunked exclusive scan (generic) =================
__global__ void __launch_bounds__(1024)
exclusive_scan_u32(const unsigned* __restrict__ in, unsigned* __restrict__ out,
                   int n, unsigned* total) {
  __shared__ unsigned sh[1024];
  __shared__ unsigned carry;
  if (threadIdx.x == 0) carry = 0u;
  __syncthreads();
  for (int base = 0; base < n; base += 1024) {
    int i = base + (int)threadIdx.x;
    unsigned v = (i < n) ? in[i] : 0u;
    sh[threadIdx.x] = v;
    __syncthreads();
    for (int off = 1; off < 1024; off <<= 1) {
      unsigned t = (threadIdx.x >= (unsigned)off) ? sh[threadIdx.x - off] : 0u;
      __syncthreads();
      sh[threadIdx.x] += t;
      __syncthreads();
    }
    if (i < n) out[i] = carry + sh[threadIdx.x] - v;   // exclusive
    __syncthreads();
    if (threadIdx.x == 1023) carry += sh[1023];
    __syncthreads();
  }
  if (threadIdx.x == 0 && total) *total = carry;
}

// ================= 6) relabel + linearized keys =================
__global__ void relabel_kernel(const int* __restrict__ cluster, const unsigned* __restrict__ newid,
                               int* clusterF, int N) {
  int v = blockIdx.x * 256 + threadIdx.x;
  if (v < N) clusterF[v] = (int)newid[cluster[v]];
}
__global__ void build_keys(const int* __restrict__ row, const int* __restrict__ col,
                           const int* __restrict__ clusterF, const unsigned* __restrict__ scal,
                           unsigned* keys, int E) {
  int e = blockIdx.x * 256 + threadIdx.x;
  if (e >= E) return;
  unsigned c = scal[0];
  keys[e] = (unsigned)clusterF[row[e]] * c + (unsigned)clusterF[col[e]];
}

// ================= 7) LSD radix sort (4 x 8-bit), stable =================
__global__ void radix_hist(const unsigned* __restrict__ keys, unsigned* hist,
                           int E, int NB, int shift) {
  __shared__ unsigned cnt[256];
  cnt[threadIdx.x] = 0u;
  __syncthreads();
  int i = blockIdx.x * 256 + threadIdx.x;
  if (i < E) atomicAdd(&cnt[(keys[i] >> shift) & 255u], 1u);
  __syncthreads();
  hist[(size_t)threadIdx.x * NB + blockIdx.x] = cnt[threadIdx.x];  // digit-major layout
}
__global__ void radix_scatter(const unsigned* __restrict__ keys, unsigned* out,
                              const unsigned* __restrict__ histScan,
                              int E, int NB, int shift) {
  __shared__ int dig[256];
  int i = blockIdx.x * 256 + threadIdx.x;
  unsigned k = 0u; int d = 300;
  if (i < E) { k = keys[i]; d = (int)((k >> shift) & 255u); }
  dig[threadIdx.x] = d;
  __syncthreads();
  if (i < E) {
    int rank = 0;
    for (int j = 0; j < (int)threadIdx.x; ++j) rank += (dig[j] == d);
    out[histScan[(size_t)d * NB + blockIdx.x] + rank] = k;
  }
}

// ================= 8) unique + output writers (device-computed offsets) =============
__global__ void uniq_flags(const unsigned* __restrict__ keys, unsigned* flags, int E) {
  int i = blockIdx.x * 256 + threadIdx.x;
  if (i >= E) return;
  flags[i] = (i == 0 || keys[i] != keys[i - 1]) ? 1u : 0u;
}
__global__ void zero_out(float* out, int n) {
  int i = blockIdx.x * 256 + threadIdx.x;
  if (i < n) out[i] = 0.f;
}
__global__ void write_edges(const unsigned* __restrict__ keys, const unsigned* __restrict__ flags,
                            const unsigned* __restrict__ pos, const unsigned* __restrict__ scal,
                            float* out, int E) {
  int i = blockIdx.x * 256 + threadIdx.x;
  if (i >= E || !flags[i]) return;
  unsigned c = scal[0], u = scal[1];
  unsigned key = keys[i], j = pos[i];
  size_t off = (size_t)c * CHN;
  out[off + j]     = (float)(key / c);
  out[off + u + j] = (float)(key % c);
}
__global__ void scatter_newx(const float* __restrict__ x, const int* __restrict__ clusterF,
                             float* out, int N) {
  int idx = blockIdx.x * 256 + threadIdx.x;
  if (idx >= N * CHN) return;
  int v = idx / CHN, ch = idx - v * CHN;
  atomicAdd(&out[(size_t)clusterF[v] * CHN + ch], x[idx]);
}
__global__ void write_score_init(const unsigned* __restrict__ scal, float* out, int N) {
  int v = blockIdx.x * 256 + threadIdx.x;
  unsigned c = scal[0], u = scal[1];
  if (v < N && (unsigned)v < c) out[(size_t)c * CHN + 2u * u + c + v] = 1.0f;
}
__global__ void write_score_match(const int* __restrict__ row, const int* __restrict__ match,
                                  const int* __restrict__ clusterF, const float* __restrict__ score,
                                  const unsigned* __restrict__ scal, float* out, int E) {
  int e = blockIdx.x * 256 + threadIdx.x;
  if (e >= E || !match[e]) return;
  unsigned c = scal[0], u = scal[1];
  out[(size_t)c * CHN + 2u * u + c + (unsigned)clusterF[row[e]]] = score[e];
}
__global__ void scale_newx(const unsigned* __restrict__ scal, float* out, int N) {
  int idx = blockIdx.x * 256 + threadIdx.x;
  unsigned c = scal[0], u = scal[1];
  if (idx >= (int)(c * CHN)) return;
  out[idx] *= out[(size_t)c * CHN + 2u * u + c + (unsigned)(idx / CHN)];
}
__global__ void write_batch(const int* __restrict__ batch, const int* __restrict__ clusterF,
                            const unsigned* __restrict__ scal, float* out, int N) {
  int v = blockIdx.x * 256 + threadIdx.x;
  if (v >= N) return;
  unsigned c = scal[0], u = scal[1];
  out[(size_t)c * CHN + 2u * u + (unsigned)clusterF[v]] = (float)batch[v];
}
__global__ void write_cluster(const int* __restrict__ clusterF, const unsigned* __restrict__ scal,
                              float* out, int N) {
  int v = blockIdx.x * 256 + threadIdx.x;
  if (v >= N) return;
  unsigned c = scal[0], u = scal[1];
  out[(size_t)c * CHN + 2u * u + 2u * c + v] = (float)clusterF[v];
}

// ==================================== launch ====================================
extern "C" void kernel_launch(void* const* d_in, const int* in_sizes, int n_in,
                              void* d_out, int out_size, void* d_ws, size_t ws_size,
                              hipStream_t stream) {
  const float* x     = (const float*)d_in[0];
  const int*   ei    = (const int*)d_in[1];
  const int*   batch = (const int*)d_in[2];
  const float* lin_w = (const float*)d_in[3];
  const float* lin_b = (const float*)d_in[4];
  const int N = in_sizes[0] / CHN;
  const int E = in_sizes[1] / 2;
  const int* row = ei;
  const int* col = ei + E;

  const int gN  = (N + 255) / 256;
  const int gE  = (E + 255) / 256;
  const int gNC = (N * CHN + 255) / 256;
  const int NB  = gE;                 // radix blocks
  const int HN  = NB * 256;           // histogram entries

  // ---- workspace carving ----
  char* wsp = (char*)d_ws;
  auto carve = [&](size_t bytes) -> void* {
    void* p = (void*)wsp;
    wsp += (bytes + 255) & ~(size_t)255;
    return p;
  };
  float*    s_lo   = (float*)carve((size_t)N * 4);
  float*    s_hi   = (float*)carve((size_t)N * 4);
  float*    raw    = (float*)carve((size_t)E * 4);
  float*    ex     = (float*)carve((size_t)E * 4);
  float*    score  = (float*)carve((size_t)E * 4);
  unsigned* nodeMax= (unsigned*)carve((size_t)N * 4);
  float*    den    = (float*)carve((size_t)N * 4);
  unsigned long long* ekey    = (unsigned long long*)carve((size_t)E * 8);
  unsigned long long* nodeKey = (unsigned long long*)carve((size_t)N * 8);
  int*      used   = (int*)carve((size_t)N * 4);
  int*      match  = (int*)carve((size_t)E * 4);
  int*      cluster  = (int*)carve((size_t)N * 4);
  int*      clusterF = (int*)carve((size_t)N * 4);
  unsigned* inSet  = (unsigned*)carve((size_t)N * 4);
  unsigned* newid  = (unsigned*)carve((size_t)N * 4);
  unsigned* keysA  = (unsigned*)carve((size_t)E * 4);
  unsigned* keysB  = (unsigned*)carve((size_t)E * 4);
  unsigned* flags  = (unsigned*)carve((size_t)E * 4);
  unsigned* pos    = (unsigned*)carve((size_t)E * 4);
  unsigned* hist     = (unsigned*)carve((size_t)HN * 4);
  unsigned* histScan = (unsigned*)carve((size_t)HN * 4);
  unsigned* scal   = (unsigned*)carve(64);    // [0]=c, [1]=u

  float* out = (float*)d_out;

  // 1) node partial scores via WMMA (f32, exact)
  const int n_tiles = N / 16;
  if (n_tiles > 0)
    node_scores_wmma<<<(n_tiles + 7) / 8, 256, 0, stream>>>(x, lin_w, s_lo, s_hi, n_tiles);
  if (N % 16)
    node_scores_tail<<<1, 256, 0, stream>>>(x, lin_w, s_lo, s_hi, n_tiles * 16, N);

  // 2-3) init + softmax edge scores
  init_nodes<<<gN, 256, 0, stream>>>(nodeMax, den, used, cluster, inSet, N);
  edge_raw_kernel<<<gE, 256, 0, stream>>>(s_lo, s_hi, lin_b, row, col, raw, nodeMax, E);
  edge_exp_kernel<<<gE, 256, 0, stream>>>(raw, col, nodeMax, ex, den, E);
  edge_score_kernel<<<gE, 256, 0, stream>>>(ex, den, col, score, ekey, match, E);

  // 4) greedy matching via locally-dominant iterations
  for (int r = 0; r < ROUNDS; ++r) {
    reset_keys<<<gN, 256, 0, stream>>>(nodeKey, N);
    match_propose<<<gE, 256, 0, stream>>>(row, col, used, ekey, nodeKey, E);
    match_accept<<<gE, 256, 0, stream>>>(row, col, used, ekey, nodeKey, match, cluster, inSet, E);
  }

  // 5-6) compact relabel: c = #representatives
  exclusive_scan_u32<<<1, 1024, 0, stream>>>(inSet, newid, N, &scal[0]);
  relabel_kernel<<<gN, 256, 0, stream>>>(cluster, newid, clusterF, N);
  build_keys<<<gE, 256, 0, stream>>>(row, col, clusterF, scal, keysA, E);

  // 7) radix sort keysA (4 passes, ping-pong A->B->A->B->A)
  unsigned* src = keysA;
  unsigned* dst = keysB;
  for (int p = 0; p < 4; ++p) {
    int shift = p * 8;
    radix_hist<<<NB, 256, 0, stream>>>(src, hist, E, NB, shift);
    exclusive_scan_u32<<<1, 1024, 0, stream>>>(hist, histScan, HN, nullptr);
    radix_scatter<<<NB, 256, 0, stream>>>(src, dst, histScan, E, NB, shift);
    unsigned* t = src; src = dst; dst = t;
  }
  // after 4 passes, sorted keys are back in keysA (== src)

  // 8) unique + outputs
  uniq_flags<<<gE, 256, 0, stream>>>(src, flags, E);
  exclusive_scan_u32<<<1, 1024, 0, stream>>>(flags, pos, E, &scal[1]);

  zero_out<<<(out_size + 255) / 256, 256, 0, stream>>>(out, out_size);
  write_edges<<<gE, 256, 0, stream>>>(src, flags, pos, scal, out, E);
  scatter_newx<<<gNC, 256, 0, stream>>>(x, clusterF, out, N);
  write_score_init<<<gN, 256, 0, stream>>>(scal, out, N);
  write_score_match<<<gE, 256, 0, stream>>>(row, match, clusterF, score, scal, out, E);
  scale_newx<<<gNC, 256, 0, stream>>>(scal, out, N);
  write_batch<<<gN, 256, 0, stream>>>(batch, clusterF, scal, out, N);
  write_cluster<<<gN, 256, 0, stream>>>(clusterF, scal, out, N);
}